// ClientBackbone_37160057045201
// MI455X (gfx1250) — compile-verified
//
#include <hip/hip_runtime.h>

typedef float v2f __attribute__((ext_vector_type(2)));
typedef float v8f __attribute__((ext_vector_type(8)));

#define BATCH 64

// ---------------------------------------------------------------------------
// 1x1 conv as GEMM per batch:  Y[b,o,s] = sum_c W[o,c] * X[b,c,s] + bias[o]
// One wave32 computes one 16(M=O) x 16(N=S) tile with V_WMMA_F32_16X16X4_F32,
// looping K(=C) in steps of 4. Grid is sized so every wave owns a tile
// (EXEC stays all-ones for WMMA). N edge handled by clamped loads + masked
// stores (only garbage columns of B feed garbage columns of D).
// ---------------------------------------------------------------------------
template <int O, int C, int S>
__global__ __launch_bounds__(256) void conv1x1_wmma_kernel(
    const float* __restrict__ X,     // (BATCH, C, S)
    const float* __restrict__ W,     // (O, C) row-major
    const float* __restrict__ bias,  // (O)
    float* __restrict__ Y)           // (BATCH, O, S)
{
  constexpr int MT = O / 16;
  constexpr int NT = (S + 15) / 16;

  const int wave = threadIdx.x >> 5;
  const int lane = threadIdx.x & 31;
  const int tile = blockIdx.x * 8 + wave;

  const int b  = tile / (MT * NT);
  const int rt = tile % (MT * NT);
  const int mt = rt / NT;
  const int nt = rt % NT;
  const int m0 = mt * 16;
  const int n0 = nt * 16;

  const float* __restrict__ Xb = X + (size_t)b * C * S;

  const int row = lane & 15;        // M-row for A / N-col for B
  const int kk  = (lane >> 4) * 2;  // K base held by this lane's regs
  const int nc  = (n0 + row < S) ? (n0 + row) : (S - 1);  // clamped load col

  // A: W[m0+row, k+kk .. k+kk+1]  (contiguous -> b64 load)
  const float* __restrict__ aptr = W + (size_t)(m0 + row) * C + kk;
  // B: Xb[(k+kk+j)*S + nc]        (stride-S between the two K regs)
  const float* __restrict__ bptr = Xb + (size_t)kk * S + nc;

  v8f acc = {0.f, 0.f, 0.f, 0.f, 0.f, 0.f, 0.f, 0.f};

#pragma unroll 4
  for (int k = 0; k < C; k += 4) {
    v2f a = *reinterpret_cast<const v2f*>(aptr + k);  // 8-byte aligned
    v2f bv;
    bv.x = bptr[(size_t)k * S];
    bv.y = bptr[(size_t)(k + 1) * S];
    acc = __builtin_amdgcn_wmma_f32_16x16x4_f32(
        /*neg_a=*/false, a, /*neg_b=*/false, bv,
        /*c_mod=*/(short)0, acc, /*reuse_a=*/false, /*reuse_b=*/false);
  }

  // C/D layout: reg r -> M = m0 + r + (lane>=16 ? 8 : 0), N = n0 + (lane&15)
  const int nOut  = n0 + row;
  const int mBase = m0 + ((lane >> 4) << 3);
  if (nOut < S) {
    float* __restrict__ yb = Y + (size_t)b * O * S + (size_t)mBase * S + nOut;
#pragma unroll
    for (int r = 0; r < 8; ++r) {
      yb[(size_t)r * S] = acc[r] + bias[mBase + r];
    }
  }
}

// ---------------------------------------------------------------------------
// 2x bilinear upsample, half-pixel centers (jax.image.resize 'bilinear').
// Output pixel (2y+dy, 2x+dx) uses input rows {y, y±1} and cols {x, x±1}
// with weights 0.75/0.25 per axis, edges clamped (== jax's normalized
// triangle weights). One thread produces the whole 2x2 output quad from the
// shared 3x3 input neighborhood: 9 loads + 2 b64 stores per 4 outputs,
// instead of 16 loads + 4 b32 stores. Outputs are never re-read -> NT stores
// keep L2 free for the conv results that ARE re-read.
// ---------------------------------------------------------------------------
struct Quad { v2f top, bot; };

__device__ __forceinline__ Quad quad2x(const float* __restrict__ p, int H,
                                       int y, int x) {
  const int ym = (y == 0) ? 0 : y - 1;
  const int yp = (y == H - 1) ? H - 1 : y + 1;
  const int xm = (x == 0) ? 0 : x - 1;
  const int xp = (x == H - 1) ? H - 1 : x + 1;

  const float c  = p[y * H + x];    // center
  const float l  = p[y * H + xm];
  const float r  = p[y * H + xp];
  const float u  = p[ym * H + x];
  const float d  = p[yp * H + x];
  const float ul = p[ym * H + xm];
  const float ur = p[ym * H + xp];
  const float dl = p[yp * H + xm];
  const float dr = p[yp * H + xp];

  Quad q;
  q.top.x = 0.5625f * c + 0.1875f * (l + u) + 0.0625f * ul;  // (2y,   2x)
  q.top.y = 0.5625f * c + 0.1875f * (r + u) + 0.0625f * ur;  // (2y,   2x+1)
  q.bot.x = 0.5625f * c + 0.1875f * (l + d) + 0.0625f * dl;  // (2y+1, 2x)
  q.bot.y = 0.5625f * c + 0.1875f * (r + d) + 0.0625f * dr;  // (2y+1, 2x+1)
  return q;
}

template <int Cn, int HL>
__global__ __launch_bounds__(256) void upsample2x_kernel(
    const float* __restrict__ in,  // (BATCH, Cn, HL, HL)
    float* __restrict__ out)       // (BATCH, Cn, 2HL, 2HL)
{
  constexpr int HO = 2 * HL;
  const size_t i = (size_t)blockIdx.x * 256 + threadIdx.x;  // one per input px
  const int x = (int)(i % HL);
  const int y = (int)((i / HL) % HL);
  const size_t bc = i / (size_t)(HL * HL);  // flattened b*Cn + c

  const float* __restrict__ p = in + bc * (size_t)(HL * HL);
  const Quad q = quad2x(p, HL, y, x);

  float* __restrict__ o = out + bc * (size_t)(HO * HO) + (size_t)(2 * y) * HO + 2 * x;
  __builtin_nontemporal_store(q.top, (v2f*)o);         // 8B aligned (2x even)
  __builtin_nontemporal_store(q.bot, (v2f*)(o + HO));
}

// aux1: upsample 28->56 fused with per-28x28-block channel permutation:
// out[b, j, h, w] = up(conv1)[b, perms[blk, j], h, w], blk = (h/28)*2 + w/28.
// Block size 28 is even -> a 2x2 quad never straddles blocks; perms lookup
// is uniform per quad.
__global__ __launch_bounds__(256) void aux1_kernel(
    const float* __restrict__ in,    // (BATCH, 64, 28, 28) conv result
    const int* __restrict__ perms,   // (784, 64), only first 4 rows used
    float* __restrict__ out)         // (BATCH, 64, 56, 56)
{
  const size_t i = (size_t)blockIdx.x * 256 + threadIdx.x;  // one per input px
  const int x = (int)(i % 28);
  const int y = (int)((i / 28) % 28);
  const int j = (int)((i / 784) % 64);
  const int b = (int)(i / (784 * 64));

  const int blk = (y / 14) * 2 + (x / 14);  // == (2y/28)*2 + (2x/28)
  const int c = perms[blk * 64 + j];

  const float* __restrict__ p = in + ((size_t)b * 64 + c) * 784;
  const Quad q = quad2x(p, 28, y, x);

  float* __restrict__ o = out + ((size_t)b * 64 + j) * 3136 + (size_t)(2 * y) * 56 + 2 * x;
  __builtin_nontemporal_store(q.top, (v2f*)o);
  __builtin_nontemporal_store(q.bot, (v2f*)(o + 56));
}

// ---------------------------------------------------------------------------
// Host launcher
// ---------------------------------------------------------------------------
extern "C" void kernel_launch(void* const* d_in, const int* in_sizes, int n_in,
                              void* d_out, int out_size, void* d_ws,
                              size_t ws_size, hipStream_t stream) {
  // Input order: f1 f2 f3 f4 x_final cw1 cb1 cw2 cb2 cw3 cb3 cw4 cb4 w1..w4 perms
  const float* f1  = (const float*)d_in[0];
  const float* f2  = (const float*)d_in[1];
  const float* f3  = (const float*)d_in[2];
  const float* f4  = (const float*)d_in[3];
  const float* xf  = (const float*)d_in[4];
  const float* cw1 = (const float*)d_in[5];
  const float* cb1 = (const float*)d_in[6];
  const float* cw2 = (const float*)d_in[7];
  const float* cb2 = (const float*)d_in[8];
  const float* cw3 = (const float*)d_in[9];
  const float* cb3 = (const float*)d_in[10];
  const float* cw4 = (const float*)d_in[11];
  const float* cb4 = (const float*)d_in[12];
  const float* w1  = (const float*)d_in[13];
  const float* w2  = (const float*)d_in[14];
  const float* w3  = (const float*)d_in[15];
  const float* w4  = (const float*)d_in[16];
  const int*   pm  = (const int*)d_in[17];

  float* out = (float*)d_out;
  // Output tuple offsets (floats), return order:
  float* o_xf = out + 0;         // (64,512)           32768
  float* o_a1 = out + 32768;     // (64,64,56,56)  12845056
  float* o_a2 = out + 12877824;  // (64,128,28,28)  6422528
  float* o_a3 = out + 19300352;  // (64,256,14,14)  3211264
  float* o_a4 = out + 22511616;  // (64,512,7,7)    1605632
  float* o_w1 = out + 24117248;  // 200704
  float* o_w2 = out + 24317952;  // 100352
  float* o_w3 = out + 24418304;  // 50176
  float* o_w4 = out + 24468480;  // 25088

  // Workspace: low-res conv results (conv commutes with per-channel upsample)
  float* ws = (float*)d_ws;
  float* c1 = ws;            // (64, 64,28,28) = 3211264 floats
  float* c2 = ws + 3211264;  // (64,128,14,14) = 1605632 floats
  float* c3 = ws + 4816896;  // (64,256, 7, 7) =  802816 floats

  // Pass-throughs (D2D async copies are graph-capture safe)
  hipMemcpyAsync(o_xf, xf, 32768 * sizeof(float), hipMemcpyDeviceToDevice, stream);
  hipMemcpyAsync(o_w1, w1, 200704 * sizeof(float), hipMemcpyDeviceToDevice, stream);
  hipMemcpyAsync(o_w2, w2, 100352 * sizeof(float), hipMemcpyDeviceToDevice, stream);
  hipMemcpyAsync(o_w3, w3, 50176 * sizeof(float), hipMemcpyDeviceToDevice, stream);
  hipMemcpyAsync(o_w4, w4, 25088 * sizeof(float), hipMemcpyDeviceToDevice, stream);

  // 1x1 convs at LOW resolution (WMMA f32). Tile counts are exact multiples
  // of 8 waves/block: no wave guard -> EXEC all-ones for every WMMA.
  conv1x1_wmma_kernel<64, 64, 784><<<(BATCH * 4 * 49) / 8, 256, 0, stream>>>(f1, cw1, cb1, c1);
  conv1x1_wmma_kernel<128, 128, 196><<<(BATCH * 8 * 13) / 8, 256, 0, stream>>>(f2, cw2, cb2, c2);
  conv1x1_wmma_kernel<256, 128, 49><<<(BATCH * 16 * 4) / 8, 256, 0, stream>>>(f3, cw3, cb3, c3);
  // aux4 has no resize (7->7 identity): GEMM straight into the output.
  conv1x1_wmma_kernel<512, 512, 49><<<(BATCH * 32 * 4) / 8, 256, 0, stream>>>(f4, cw4, cb4, o_a4);

  // Upsample (+ block-wise channel shuffle for aux1) writebacks.
  // One thread per INPUT pixel -> 2x2 output quad; grids are exact /256.
  aux1_kernel<<<3211264 / 256, 256, 0, stream>>>(c1, pm, o_a1);
  upsample2x_kernel<128, 14><<<1605632 / 256, 256, 0, stream>>>(c2, o_a2);
  upsample2x_kernel<256, 7><<<802816 / 256, 256, 0, stream>>>(c3, o_a3);
}